// LorentzConv1d_14302241096029
// MI455X (gfx1250) — compile-verified
//
#include <hip/hip_runtime.h>
#include <math.h>

typedef __attribute__((ext_vector_type(2))) float v2f;
typedef __attribute__((ext_vector_type(8))) float v8f;

#define L_LEN     16384
#define NCH_SPACE 15
#define KSIZE     9
#define PAD       4
#define DCAT      136            // 1 + 15*9
#define OUTC      64             // 1 (time) + 63 (h)
#define PW        128            // patches per workgroup
#define XW        (PW + 2*PAD)   // 136 staged positions
#define WPITCH    138            // even (b64-aligned pairs), stride 10 mod 64 -> conflict-free
#define HPITCH    129            // stride 1 mod 64 -> conflict-free scatter

// smem layout (float offsets); W region is reused for H after GEMM.
#define X_OFF     0                        // 15*136 = 2040
#define TIME_OFF  (X_OFF + NCH_SPACE*XW)   // 2040 .. 2167
#define B_OFF     (TIME_OFF + PW)          // 2168 .. 2231
#define W_OFF     2176                     // even; 64*138 = 8832 (>= 64*129 for H)
#define SMEM_FLOATS (W_OFF + OUTC*WPITCH)  // 11008 floats = 44 KB

// LDS float-offset of concat feature d, relative to a patch-row base.
__device__ constexpr int featOff(int d) {
    return (d == 0) ? TIME_OFF
                    : (X_OFF + ((d - 1) % NCH_SPACE) * XW + ((d - 1) / NCH_SPACE));
}

__global__ __launch_bounds__(128)
void lorentz_conv1d_wmma(const float* __restrict__ x,
                         const float* __restrict__ W,
                         const float* __restrict__ bias,
                         float* __restrict__ out)
{
    __shared__ float smem[SMEM_FLOATS];

    const int tid  = threadIdx.x;
    const int blk  = blockIdx.x;
    const int b    = blk >> 7;             // 16384/128 = 128 blocks per batch
    const int pblk = blk & 127;
    const int p0   = pblk * PW;

    // ---- stage W (row 63 zeroed, pitch 138) + bias ----
    for (int i = tid; i < OUTC * DCAT; i += 128) {
        int o = i / DCAT;
        int d = i - o * DCAT;
        smem[W_OFF + o * WPITCH + d] = (o < OUTC - 1) ? W[o * DCAT + d] : 0.0f;
    }
    if (tid < OUTC) smem[B_OFF + tid] = (tid < OUTC - 1) ? bias[tid] : 0.0f;

    // ---- stage x space-channel window (zero pad at sequence edges) ----
    for (int i = tid; i < NCH_SPACE * XW; i += 128) {
        int ch = i / XW;
        int j  = i - ch * XW;
        int gp = p0 - PAD + j;
        float v = 0.0f;
        if (gp >= 0 && gp < L_LEN)
            v = x[((size_t)(b * 16 + 1 + ch)) * L_LEN + gp];
        smem[X_OFF + ch * XW + j] = v;
    }
    __syncthreads();

    // ---- per-patch Lorentz time feature: sqrt(1 + sum space^2) ----
    {
        float ss = 0.0f;
        for (int ch = 0; ch < NCH_SPACE; ++ch) {
            const float* row = &smem[X_OFF + ch * XW + tid];
            #pragma unroll
            for (int tap = 0; tap < KSIZE; ++tap) {
                float v = row[tap];
                ss += v * v;
            }
        }
        smem[TIME_OFF + tid] = sqrtf(1.0f + ss);
    }
    __syncthreads();

    // ---- GEMM: [2 x 16 patches x 136] x [136 x 64] via v_wmma_f32_16x16x4_f32 ----
    const int wave  = tid >> 5;            // 0..3
    const int lane  = tid & 31;
    const int half  = lane >> 4;           // selects K pair within a step
    const int sub   = lane & 15;           // A: row / B: column within tile
    const int prow0 = wave * 16 + sub;     // first M tile
    const int prow1 = prow0 + 64;          // second M tile (reuses B)
    const float* wbase = &smem[W_OFF + sub * WPITCH + half * 2];

    v8f acc[2][4] = {};

    #pragma unroll
    for (int kt = 0; kt < DCAT / 4; ++kt) {        // 34 fully-unrolled K-steps
        // compile-time feature offsets for both lane halves
        const int oxlo = featOff(kt * 4 + 0);
        const int oylo = featOff(kt * 4 + 1);
        const int oxhi = featOff(kt * 4 + 2);
        const int oyhi = featOff(kt * 4 + 3);
        const int ox = half ? oxhi : oxlo;         // one v_cndmask each
        const int oy = half ? oyhi : oylo;

        v2f a0, a1;
        a0.x = smem[ox + prow0];
        a0.y = smem[oy + prow0];
        a1.x = smem[ox + prow1];
        a1.y = smem[oy + prow1];

        #pragma unroll
        for (int t = 0; t < 4; ++t) {
            v2f bb;
            bb.x = wbase[t * 16 * WPITCH + kt * 4];
            bb.y = wbase[t * 16 * WPITCH + kt * 4 + 1];
            acc[0][t] = __builtin_amdgcn_wmma_f32_16x16x4_f32(
                false, a0, false, bb, (short)0, acc[0][t], false, false);
            acc[1][t] = __builtin_amdgcn_wmma_f32_16x16x4_f32(
                false, a1, false, bb, (short)0, acc[1][t], false, false);
        }
    }

    __syncthreads();                  // all waves done reading W -> reuse its LDS for H
    float* sH = &smem[W_OFF];

    // ---- bias + ReLU, scatter C-fragments into sH (conflict-free pitch) ----
    #pragma unroll
    for (int mt = 0; mt < 2; ++mt) {
        const int pbase = wave * 16 + mt * 64;
        #pragma unroll
        for (int t = 0; t < 4; ++t) {
            int o = t * 16 + sub;             // h channel 0..63 (63 = pad, dropped)
            if (o < OUTC - 1) {
                float bv = smem[B_OFF + o];
                #pragma unroll
                for (int g = 0; g < 8; ++g) {
                    float hv = fmaxf(acc[mt][t][g] + bv, 0.0f);
                    sH[(1 + o) * HPITCH + pbase + g + half * 8] = hv;
                }
            }
        }
    }
    __syncthreads();

    // ---- output time channel: sqrt(1 + sum h^2) ----
    {
        float ss = 0.0f;
        for (int o = 1; o < OUTC; ++o) {
            float v = sH[o * HPITCH + tid];
            ss += v * v;
        }
        sH[tid] = sqrtf(1.0f + ss);           // row 0
    }
    __syncthreads();

    // ---- coalesced store: out[b, och, p0..p0+127] ----
    const size_t obase = ((size_t)b * OUTC) * L_LEN + (size_t)p0;
    for (int i = tid; i < OUTC * PW; i += 128) {
        int och = i >> 7;
        int p   = i & 127;
        out[obase + (size_t)och * L_LEN + p] = sH[och * HPITCH + p];
    }
}

extern "C" void kernel_launch(void* const* d_in, const int* in_sizes, int n_in,
                              void* d_out, int out_size, void* d_ws, size_t ws_size,
                              hipStream_t stream) {
    const float* x  = (const float*)d_in[0];   // (32, 16, 16384) f32
    const float* W  = (const float*)d_in[1];   // (63, 136) f32
    const float* bb = (const float*)d_in[2];   // (63,) f32
    float* out = (float*)d_out;                // (32, 64, 16384) f32

    dim3 grid(32 * (L_LEN / PW));              // 4096 workgroups
    dim3 block(128);                           // 4 waves, two 16-patch M tiles each
    hipLaunchKernelGGL(lorentz_conv1d_wmma, grid, block, 0, stream, x, W, bb, out);
}